// GlobalAverageBlock_35725537968381
// MI455X (gfx1250) — compile-verified
//
#include <hip/hip_runtime.h>

// ---------------------------------------------------------------------------
// Segment-mean (B=16 segments, D=256 features) via CDNA5 WMMA.
//   sums = S^T * X with one-hot selector S, done with V_WMMA_F32_16X16X4_F32
//   (full f32 precision; M=16 == number of segments).
// Memory-bound: 1 GB streamed once -> 2-deep pipelined NT loads, partial
// s_wait_loadcnt, one WMMA per 4 rows.
// ---------------------------------------------------------------------------

typedef __attribute__((ext_vector_type(2))) float v2f;
typedef __attribute__((ext_vector_type(8))) float v8f;
typedef __attribute__((ext_vector_type(2))) int   v2i;

#define SEGS      16
#define DIM       256
#define ROWBLOCKS 64   // grid.y for the sum kernel

__global__ __launch_bounds__(256)
void GlobalAverageBlock_zero_ws(float* __restrict__ p) {
    p[blockIdx.x * 256 + threadIdx.x] = 0.0f;
}

__device__ __forceinline__ v8f gab_wmma(int l, int id0, int id1,
                                        float b0, float b1, v8f acc) {
    v2f A;  A.x  = (id0 == l) ? 1.0f : 0.0f;  A.y  = (id1 == l) ? 1.0f : 0.0f;
    v2f Bv; Bv.x = b0;                        Bv.y = b1;
    // (neg_a, A, neg_b, B, c_mod, C, reuse_a, reuse_b)
    return __builtin_amdgcn_wmma_f32_16x16x4_f32(
        false, A, false, Bv, (short)0, acc, false, false);
}

__global__ __launch_bounds__(256)
void GlobalAverageBlock_seg_sum_wmma(const float* __restrict__ x,
                                     const int*   __restrict__ ids,
                                     float*       __restrict__ partial, // [SEGS*DIM]
                                     int n)
{
    __shared__ float lds[8 * 256];

    const int lane = threadIdx.x & 31;
    const int wave = threadIdx.x >> 5;            // 0..7
    const int col0 = blockIdx.x * 16;             // 16-wide feature tile

    // Row range for this wave (wave-uniform -> EXEC stays all-1s for WMMA).
    const int wavesPerTile = gridDim.y * 8;
    const int waveId       = blockIdx.y * 8 + wave;
    int rowsPerWave = (((n + wavesPerTile - 1) / wavesPerTile) + 3) & ~3;
    int r0 = waveId * rowsPerWave;
    int r1 = r0 + rowsPerWave; if (r1 > n) r1 = n;

    // ISA lane decomposition for 32-bit 16x4 A / 4x16 B operands:
    //   lanes 0-15  -> K = {0,1} in VGPR {0,1}
    //   lanes 16-31 -> K = {2,3} in VGPR {0,1}
    const int half = lane >> 4;
    const int l    = lane & 15;     // A: M index (segment); B: N index (column)
    const int k0   = half * 2;      // row offsets handled by this lane: k0, k0+1

    v8f acc = {};
    const int total = (r1 > r0) ? (r1 - r0) : 0;
    const int iters = total >> 2;           // full 4-row groups, all rows valid

    if (iters > 0) {
        const float* px = x + (size_t)(r0 + k0) * DIM + col0 + l;
        const int*   pi = ids + r0 + k0;    // even index -> 8B aligned pair

        // --- 2-deep software pipeline: groups j (a_*) and j+1 (b_*) in
        //     flight while WMMA consumes group j-? ; keeps ~6 vmem ops
        //     outstanding so waits stay partial. ---
        float a_b0 = __builtin_nontemporal_load(px);
        float a_b1 = __builtin_nontemporal_load(px + DIM);
        v2i   a_id = *(const v2i*)pi;

        if (iters > 1) {
            px += 4 * DIM; pi += 4;
            float b_b0 = __builtin_nontemporal_load(px);
            float b_b1 = __builtin_nontemporal_load(px + DIM);
            v2i   b_id = *(const v2i*)pi;

            #pragma unroll 2
            for (int i = 2; i < iters; ++i) {
                px += 4 * DIM; pi += 4;
                float c_b0 = __builtin_nontemporal_load(px);
                float c_b1 = __builtin_nontemporal_load(px + DIM);
                v2i   c_id = *(const v2i*)pi;

                acc = gab_wmma(l, a_id.x, a_id.y, a_b0, a_b1, acc);

                a_b0 = b_b0; a_b1 = b_b1; a_id = b_id;
                b_b0 = c_b0; b_b1 = c_b1; b_id = c_id;
            }
            acc = gab_wmma(l, a_id.x, a_id.y, a_b0, a_b1, acc);
            a_b0 = b_b0; a_b1 = b_b1; a_id = b_id;
        }
        acc = gab_wmma(l, a_id.x, a_id.y, a_b0, a_b1, acc);   // drain
    }

    // --- tail: at most 3 rows, branchless inside a wave-uniform guard ---
    if (total & 3) {
        const int rt   = r0 + (total & ~3);
        const int row0 = rt + k0;
        const int row1 = row0 + 1;
        // Clamp load index for safety; invalid rows get id=-1, which zeroes
        // their one-hot A column so the (finite) garbage B value is ignored.
        const int rc0 = (row0 < n) ? row0 : (n - 1);
        const int rc1 = (row1 < n) ? row1 : (n - 1);
        float b0 = x[(size_t)rc0 * DIM + col0 + l];
        float b1 = x[(size_t)rc1 * DIM + col0 + l];
        int  id0 = (row0 < r1) ? ids[rc0] : -1;
        int  id1 = (row1 < r1) ? ids[rc1] : -1;
        acc = gab_wmma(l, id0, id1, b0, b1, acc);
    }

    // acc layout: VGPR g, lanes 0-15 -> seg=g,   col=l
    //                     lanes 16-31 -> seg=8+g, col=l
    #pragma unroll
    for (int g = 0; g < 8; ++g) {
        int seg = g + 8 * half;
        lds[wave * 256 + seg * 16 + l] = acc[g];
    }
    __syncthreads();

    // Cross-wave reduce (8 waves -> 1 tile), one f32 atomic per element.
    const int t = threadIdx.x;          // 0..255 == seg*16 + col
    float s = 0.0f;
    #pragma unroll
    for (int w = 0; w < 8; ++w) s += lds[w * 256 + t];
    const int seg = t >> 4;
    const int col = t & 15;
    unsafeAtomicAdd(&partial[seg * DIM + col0 + col], s);
}

__device__ __forceinline__ int gab_lower_bound(const int* __restrict__ ids, int n, int v) {
    int lo = 0, hi = n;
    while (lo < hi) {
        int mid = (lo + hi) >> 1;
        if (ids[mid] < v) lo = mid + 1; else hi = mid;
    }
    return lo;
}

__global__ __launch_bounds__(256)
void GlobalAverageBlock_finalize(const float* __restrict__ partial,
                                 const int*   __restrict__ ids,
                                 float*       __restrict__ out,
                                 int n)
{
    __shared__ int bounds[SEGS + 1];
    const int t = threadIdx.x;
    if (t < SEGS + 1) bounds[t] = gab_lower_bound(ids, n, t);  // sorted ids -> counts
    __syncthreads();

    #pragma unroll
    for (int seg = 0; seg < SEGS; ++seg) {
        float cnt   = (float)(bounds[seg + 1] - bounds[seg]);
        float denom = cnt > 1.0f ? cnt : 1.0f;                 // max(counts, 1)
        out[seg * DIM + t] = partial[seg * DIM + t] / denom;
    }
}

extern "C" void kernel_launch(void* const* d_in, const int* in_sizes, int n_in,
                              void* d_out, int out_size, void* d_ws, size_t ws_size,
                              hipStream_t stream) {
    const float* x   = (const float*)d_in[0];   // [N, 256] f32
    const int*   ids = (const int*)  d_in[1];   // [N] sorted int32
    float*       out = (float*)d_out;           // [16, 256] f32
    float*       ws  = (float*)d_ws;            // 16*256 f32 accumulator (16 KB)
    const int    n   = in_sizes[1];             // N (rows)

    GlobalAverageBlock_zero_ws<<<dim3(SEGS), dim3(256), 0, stream>>>(ws);
    GlobalAverageBlock_seg_sum_wmma<<<dim3(DIM / 16, ROWBLOCKS), dim3(256), 0, stream>>>(
        x, ids, ws, n);
    GlobalAverageBlock_finalize<<<dim3(1), dim3(256), 0, stream>>>(ws, ids, out, n);
}